// focal_loss_65111704207799
// MI455X (gfx1250) — compile-verified
//
#include <hip/hip_runtime.h>
#include <stdint.h>

#define T_STEPS   256
#define N_BATCH   64
#define C_CLASSES 7356
#define S_MAX     32
#define L_EXT     (2 * S_MAX + 1)   // 65 extended states
#define NEGINF    (-1000000000.0f)

// ---- CDNA5 async global->LDS gather helpers -------------------------------
// Flat pointers to __shared__ carry the LDS offset in their low 32 bits
// (flat->LDS aperture mapping, CDNA5 ISA 10.2).
__device__ __forceinline__ uint32_t lds_off(const void* p) {
  return (uint32_t)(uintptr_t)p;
}

// Per-lane async DMA: each lane copies 4 bytes from the global address in vaddr to the LDS address in vdst; tracked by ASYNCcnt.
__device__ __forceinline__ void async_gather_f32(uint32_t ldsaddr, const float* g) {
  uint64_t ga = (uint64_t)(uintptr_t)g;
  asm volatile("global_load_async_to_lds_b32 %0, %1, off"
               :
               : "v"(ldsaddr), "v"(ga)
               : "memory");
}

__device__ __forceinline__ void wait_async_le1() {
#if __has_builtin(__builtin_amdgcn_s_wait_asynccnt)
  __builtin_amdgcn_s_wait_asynccnt(1);
#else
  asm volatile("s_wait_asynccnt 1" ::: "memory");
#endif
}
__device__ __forceinline__ void wait_async_0() {
#if __has_builtin(__builtin_amdgcn_s_wait_asynccnt)
  __builtin_amdgcn_s_wait_asynccnt(0);
#else
  asm volatile("s_wait_asynccnt 0" ::: "memory");
#endif
}

// ---- numerics -------------------------------------------------------------
__device__ __forceinline__ float lae(float a, float b) {
  float m = fmaxf(a, b);
  float d = fminf(a, b) - m;          // <= 0; underflows to 0 for ~NEGINF gaps
  return m + log1pf(__expf(d));
}

// ---- kernel 1: per-batch CTC alpha recursion ------------------------------
// grid = N_BATCH blocks, block = 96 threads (3 waves); thread l owns state l.
__global__ void __launch_bounds__(96)
ctc_alpha_kernel(const float* __restrict__ logp,   // (T, N, C) log-softmax
                 const int*   __restrict__ text,   // (N, S)
                 const int*   __restrict__ pre_len,// (N,)
                 const int*   __restrict__ lengths,// (N,)
                 float*       __restrict__ per_out)// (N,) scratch
{
  const int n = blockIdx.x;
  const int l = threadIdx.x;

  __shared__ float alpha[2][L_EXT + 2]; // [0],[1] = NEG pads -> branch-free l-1,l-2
  __shared__ float lp[2][L_EXT];        // double-buffered gathered log-probs
  __shared__ float endv[2];

  const int prelast = pre_len[n] - 1;
  const int tlen    = lengths[n];

  // Per-thread extended label + skip flag (registers only).
  int  lab  = 0;
  bool skip = false;
  if (l < L_EXT) {
    if (l & 1) lab = text[n * S_MAX + ((l - 1) >> 1)];
    if ((l & 1) && l >= 3)
      skip = (lab != text[n * S_MAX + ((l - 3) >> 1)]);
    // even l: blank (lab==0) -> skip stays false; l==1 -> false
  }
  if (l < 2) { alpha[0][l] = NEGINF; alpha[1][l] = NEGINF; }

  const float* gbase   = logp + (size_t)n * C_CLASSES + lab; // t = 0 address
  const size_t tstride = (size_t)N_BATCH * C_CLASSES;        // floats per t

  // t = 0: gather, init alpha0
  if (l < L_EXT) async_gather_f32(lds_off(&lp[0][l]), gbase);
  wait_async_0();
  if (l < L_EXT) alpha[0][2 + l] = (l < 2) ? lp[0][l] : NEGINF;
  __syncthreads();
  if (l == 0 && prelast == 0) {
    endv[0] = alpha[0][2 + 2 * tlen];
    endv[1] = alpha[0][2 + 2 * tlen - 1];
  }
  // prefetch t = 1
  if (l < L_EXT) async_gather_f32(lds_off(&lp[1][l]), gbase + tstride);

  // running prefetch address: at iteration t we issue the gather for t+1
  const float* gpref = gbase + 2 * tstride;

  #pragma unroll 2
  for (int t = 1; t < T_STEPS; ++t) {
    const int cb = t & 1, pb = cb ^ 1;
    if (t + 1 < T_STEPS) {
      if (l < L_EXT) async_gather_f32(lds_off(&lp[pb][l]), gpref);
      wait_async_le1();     // step-t gather (in-order) has landed in LDS
    } else {
      wait_async_0();
    }
    gpref += tstride;
    if (l < L_EXT) {
      float a0 = alpha[pb][2 + l];                  // alpha[l]
      float a1 = alpha[pb][1 + l];                  // alpha[l-1] (pad-safe)
      float a2 = skip ? alpha[pb][l] : NEGINF;      // alpha[l-2] if allowed
      float v  = lae(lae(a0, a1), a2);
      alpha[cb][2 + l] = v + lp[cb][l];
    }
    __syncthreads();
    if (l == 0 && t == prelast) {
      endv[0] = alpha[cb][2 + 2 * tlen];
      endv[1] = alpha[cb][2 + 2 * tlen - 1];
    }
  }

  if (l == 0) {
    float loglike = lae(endv[0], endv[1]);
    per_out[n] = -loglike / (float)tlen;
  }
}

// ---- kernel 2: mean over batch + focal weighting --------------------------
__global__ void __launch_bounds__(64)
ctc_finalize_kernel(const float* __restrict__ per, float* __restrict__ out) {
  __shared__ float s[N_BATCH];
  const int i = threadIdx.x;
  s[i] = per[i];
  __syncthreads();
  if (i == 0) {
    float acc = 0.f;
    #pragma unroll
    for (int k = 0; k < N_BATCH; ++k) acc += s[k];
    float loss = acc / (float)N_BATCH;
    float w = 1.0f - __expf(-loss);
    out[0] = w * w * loss;
  }
}

// ---- host entry -----------------------------------------------------------
extern "C" void kernel_launch(void* const* d_in, const int* in_sizes, int n_in,
                              void* d_out, int out_size, void* d_ws, size_t ws_size,
                              hipStream_t stream) {
  const float* logp   = (const float*)d_in[0];   // (T, N, C) f32
  const int*   text   = (const int*)  d_in[1];   // (N, S) i32
  const int*   prelen = (const int*)  d_in[2];   // (N,) i32
  const int*   lens   = (const int*)  d_in[3];   // (N,) i32
  float*       per    = (float*)d_ws;            // N floats scratch
  float*       out    = (float*)d_out;           // scalar

  ctc_alpha_kernel<<<N_BATCH, 96, 0, stream>>>(logp, text, prelen, lens, per);
  ctc_finalize_kernel<<<1, N_BATCH, 0, stream>>>(per, out);
}